// RNN_84052509983268
// MI455X (gfx1250) — compile-verified
//
#include <hip/hip_runtime.h>
#include <hip/hip_bf16.h>
#include <math.h>

// Problem constants (match reference)
#define B_ROWS   4096
#define SEQ      2048
#define T_CHUNK  64                  // timesteps staged per async chunk
#define N_CHUNK  (SEQ / T_CHUNK)     // 32 chunks
#define ROW_PAD  68                  // 64 data + 4 pad floats: rows stay 16B-aligned,
                                     // ds_load_b128 bank conflicts limited to 2-way
#define BLOCK    128                 // 4 wave32s per block
#define WAVES_PB (BLOCK / 32)
#define GRID     (B_ROWS / BLOCK)    // 32 blocks -> 128 waves total

// ---- CDNA5 hardware tanh (TRANS pipe) --------------------------------------
#if __has_builtin(__builtin_amdgcn_tanhf)
#define TANHF(v) __builtin_amdgcn_tanhf(v)
#else
#define TANHF(v) tanhf(v)
#endif

// ---- async global->LDS copy (ASYNCcnt path), raw CDNA5 asm ------------------
// GLOBAL_LOAD_ASYNC_TO_LDS_B128 (opcode 98, CDNA5 ISA):
//   VDST  = VGPR holding LDS byte offset (plus the immediate offset)
//   VADDR = VGPR pair holding 64-bit global address (plus the immediate offset)
// IOFFSET is added to BOTH sides (15.18 pseudocode), so a whole chunk shares one
// address pair + one LDS-offset VGPR; the 16-byte steps ride in the immediate.
template <int I>
struct AsyncChunk {
  static __device__ __forceinline__ void run(const float* g, unsigned int loff) {
    AsyncChunk<I - 1>::run(g, loff);
    asm volatile("global_load_async_to_lds_b128 %0, %1, off offset:%2"
                 :: "v"(loff), "v"(g), "i"((I - 1) * 16) : "memory");
  }
};
template <>
struct AsyncChunk<0> {
  static __device__ __forceinline__ void run(const float*, unsigned int) {}
};

static __device__ __forceinline__ void async_copy_chunk(const float* gsrc, const float* ldst) {
  // Generic pointers to __shared__ carry the LDS byte offset in their low 32
  // bits (aperture layout, CDNA5 ISA).
  unsigned int loff = (unsigned int)(unsigned long long)ldst;
  AsyncChunk<16>::run(gsrc, loff);   // 16 x b128 = 64 floats of this lane's row
}

// SOPP wait on ASYNCcnt (opcode 74).  Memory clobber fences compiler ordering
// so tile ds_loads cannot be hoisted above the wait.
#define WAIT_ASYNC(n) asm volatile("s_wait_asynccnt %0" :: "i"(n) : "memory")

// ---------------------------------------------------------------------------
// One lane owns one batch row's entire recurrence.  x is DMA'd through LDS in
// 64-step chunks, double buffered: while chunk c is consumed, chunk c+1 is in
// flight on the async queue (16 b128 descriptors per lane per chunk; async
// loads complete in order, so s_wait_asynccnt 16 drains exactly the previous
// chunk while the next chunk's 16 remain outstanding).
// ---------------------------------------------------------------------------
__global__ __launch_bounds__(BLOCK) void rnn_scan_kernel(
    const float* __restrict__ x,     // [B, S, 1]
    const float* __restrict__ W_ih,  // [3, 1]
    const float* __restrict__ b_ih,  // [3]
    const float* __restrict__ W_hh,  // [3, 3]
    const float* __restrict__ b_hh,  // [3]
    const float* __restrict__ W_out, // [1, 3]
    const float* __restrict__ b_out, // [1]
    float* __restrict__ out)         // [B, 1]
{
  __shared__ float smem[WAVES_PB][2][32 * ROW_PAD];

  const int tid  = threadIdx.x;
  const int wave = tid >> 5;
  const int lane = tid & 31;
  const int row  = blockIdx.x * BLOCK + tid;   // B divides evenly: EXEC all-ones

  // Wave-uniform weights -> scalar loads / SGPRs.
  const float wih0 = W_ih[0], wih1 = W_ih[1], wih2 = W_ih[2];
  const float bi0  = b_ih[0] + b_hh[0];
  const float bi1  = b_ih[1] + b_hh[1];
  const float bi2  = b_ih[2] + b_hh[2];
  const float w00 = W_hh[0], w01 = W_hh[1], w02 = W_hh[2];
  const float w10 = W_hh[3], w11 = W_hh[4], w12 = W_hh[5];
  const float w20 = W_hh[6], w21 = W_hh[7], w22 = W_hh[8];
  const float wo0 = W_out[0], wo1 = W_out[1], wo2 = W_out[2];
  const float bo  = b_out[0];

  const float* xrow = x + (size_t)row * SEQ;

  // Prologue: DMA chunk 0 into buffer 0.
  async_copy_chunk(xrow, &smem[wave][0][lane * ROW_PAD]);

  float h0 = 0.f, h1 = 0.f, h2 = 0.f;
  int buf = 0;

#pragma unroll 1
  for (int c = 0; c < N_CHUNK; ++c) {
    if (c + 1 < N_CHUNK) {
      // Kick off next chunk into the other buffer, then drain only the
      // previous chunk's 16 descriptors.
      async_copy_chunk(xrow + (c + 1) * T_CHUNK,
                       &smem[wave][buf ^ 1][lane * ROW_PAD]);
      WAIT_ASYNC(16);
    } else {
      WAIT_ASYNC(0);
    }

    const float* tile = &smem[wave][buf][lane * ROW_PAD];

#pragma unroll
    for (int t = 0; t < T_CHUNK; t += 4) {
      // ds_load_b128: 16B-aligned (ROW_PAD*4 = 272 = 17*16).
      const float4 xv = *reinterpret_cast<const float4*>(tile + t);
      const float xs[4] = {xv.x, xv.y, xv.z, xv.w};
#pragma unroll
      for (int k = 0; k < 4; ++k) {
        // Input projection: independent of h -> overlaps prior step's tanh.
        const float p0 = fmaf(xs[k], wih0, bi0);
        const float p1 = fmaf(xs[k], wih1, bi1);
        const float p2 = fmaf(xs[k], wih2, bi2);
        // a_j = p_j + h @ W_hh[j, :]  (3-deep FMA chain)
        const float a0 = fmaf(h2, w02, fmaf(h1, w01, fmaf(h0, w00, p0)));
        const float a1 = fmaf(h2, w12, fmaf(h1, w11, fmaf(h0, w10, p1)));
        const float a2 = fmaf(h2, w22, fmaf(h1, w21, fmaf(h0, w20, p2)));
        h0 = TANHF(a0);
        h1 = TANHF(a1);
        h2 = TANHF(a2);
      }
    }
    buf ^= 1;
  }

  // Final linear 3 -> 1 on the last hidden state.
  out[row] = fmaf(h2, wo2, fmaf(h1, wo1, fmaf(h0, wo0, bo)));
}

extern "C" void kernel_launch(void* const* d_in, const int* in_sizes, int n_in,
                              void* d_out, int out_size, void* d_ws, size_t ws_size,
                              hipStream_t stream) {
  (void)in_sizes; (void)n_in; (void)out_size; (void)d_ws; (void)ws_size;
  const float* x     = (const float*)d_in[0];
  const float* W_ih  = (const float*)d_in[1];
  const float* b_ih  = (const float*)d_in[2];
  const float* W_hh  = (const float*)d_in[3];
  const float* b_hh  = (const float*)d_in[4];
  const float* W_out = (const float*)d_in[5];
  const float* b_out = (const float*)d_in[6];

  rnn_scan_kernel<<<GRID, BLOCK, 0, stream>>>(
      x, W_ih, b_ih, W_hh, b_hh, W_out, b_out, (float*)d_out);
}